// InformationPrototype_59038620451265
// MI455X (gfx1250) — compile-verified
//
#include <hip/hip_runtime.h>
#include <hip/hip_bf16.h>
#include <math.h>

typedef __attribute__((ext_vector_type(2))) float v2f;
typedef __attribute__((ext_vector_type(4))) float v4f;
typedef __attribute__((ext_vector_type(8))) float v8f;

#define N_ROWS 4096
#define DIM    256
#define N_CLS  1000
#define EPS_CS 1e-8f

// ---------------- kernel 1: zero prototype sums + counts ----------------
__global__ void k_zero(float* __restrict__ proto, unsigned* __restrict__ counts) {
  int i = blockIdx.x * blockDim.x + threadIdx.x;
  if (i < N_CLS * DIM) proto[i] = 0.0f;
  int j = i - N_CLS * DIM;
  if (j >= 0 && j < N_CLS) counts[j] = 0u;
}

// ---------------- kernel 2: argmax(logits) per row (== argmax softmax) --
__global__ void k_argmax(const float* __restrict__ logits,
                         int* __restrict__ cls, unsigned* __restrict__ counts) {
  int gid  = blockIdx.x * blockDim.x + threadIdx.x;
  int row  = gid >> 5;
  int lane = gid & 31;
  if (row >= N_ROWS) return;
  const float* p = logits + (size_t)row * N_CLS;
  float best = -INFINITY;
  int   bi   = N_CLS;
  for (int k = lane; k < N_CLS; k += 32) {
    float v = p[k];
    if (v > best) { best = v; bi = k; }     // strided ascending k: first-max per lane
  }
#pragma unroll
  for (int off = 16; off > 0; off >>= 1) {
    float ov = __shfl_xor(best, off, 32);
    int   oi = __shfl_xor(bi,   off, 32);
    if (ov > best || (ov == best && oi < bi)) { best = ov; bi = oi; }
  }
  if (lane == 0) { cls[row] = bi; atomicAdd(&counts[bi], 1u); }
}

// ---------------- kernel 3: row L2 norms --------------------------------
__global__ void k_rownorm(const float* __restrict__ x, float* __restrict__ norms) {
  int gid  = blockIdx.x * blockDim.x + threadIdx.x;
  int row  = gid >> 5;
  int lane = gid & 31;
  if (row >= N_ROWS) return;
  const v4f* p = (const v4f*)(x + (size_t)row * DIM);
  v4f a = p[lane];
  v4f b = p[lane + 32];
  float s = a.x*a.x + a.y*a.y + a.z*a.z + a.w*a.w
          + b.x*b.x + b.y*b.y + b.z*b.z + b.w*b.w;
#pragma unroll
  for (int off = 16; off > 0; off >>= 1) s += __shfl_xor(s, off, 32);
  if (lane == 0) norms[row] = sqrtf(s);
}

// ---------------- kernel 4: fused cos-sim GEMM + aggregation + scatter --
__device__ __forceinline__ v8f wmma4(v2f a, v2f b, v8f c) {
  // D(16x16,f32) = A(16x4,f32) * B(4x16,f32) + C
  return __builtin_amdgcn_wmma_f32_16x16x4_f32(false, a, false, b, (short)0, c,
                                               false, false);
}

__launch_bounds__(64, 1)
__global__ void k_fused_agg(const float* __restrict__ x,
                            const float* __restrict__ norms,
                            const int*   __restrict__ cls,
                            float*       __restrict__ proto_sums) {
  __shared__ float Xi[2][16][260];   // per-wave 16 rows of X, padded stride
  __shared__ float XjT[DIM][18];     // transposed j-chunk: [channel][jrow]
  __shared__ float Ssc[2][16][18];   // scaled similarity tile, per wave
  __shared__ float normI[32];
  __shared__ float normJ[16];

  const int tid  = threadIdx.x;
  const int wave = tid >> 5;
  const int lane = tid & 31;
  const int half = lane >> 4;        // 0 | 1
  const int l16  = lane & 15;
  const int i0   = blockIdx.x * 32;  // 32 output rows per block (16 per wave)

  // stage Xi (32 x 256) once
  for (int idx = tid; idx < 32 * (DIM / 4); idx += 64) {
    int r  = idx >> 6;               // DIM/4 == 64
    int c4 = idx & 63;
    v4f v = *(const v4f*)(x + (size_t)(i0 + r) * DIM + c4 * 4);
    float* dst = &Xi[r >> 4][r & 15][c4 * 4];
    dst[0] = v.x; dst[1] = v.y; dst[2] = v.z; dst[3] = v.w;
  }
  if (tid < 32) normI[tid] = norms[i0 + tid];
  __syncthreads();

  float nIm[8];                      // row norms matching C/D layout rows
#pragma unroll
  for (int r = 0; r < 8; ++r) nIm[r] = normI[wave * 16 + r + 8 * half];

  v8f zero = {};
  v8f acc[16];
#pragma unroll
  for (int t = 0; t < 16; ++t) acc[t] = zero;

  for (int j0 = 0; j0 < N_ROWS; j0 += 16) {
    __syncthreads();                 // previous chunk fully consumed
    // stage Xj chunk transposed: XjT[channel][jrow]
    for (int idx = tid; idx < 16 * (DIM / 4); idx += 64) {
      int jr = idx >> 6;
      int c4 = idx & 63;
      v4f v = *(const v4f*)(x + (size_t)(j0 + jr) * DIM + c4 * 4);
      XjT[c4 * 4 + 0][jr] = v.x;
      XjT[c4 * 4 + 1][jr] = v.y;
      XjT[c4 * 4 + 2][jr] = v.z;
      XjT[c4 * 4 + 3][jr] = v.w;
    }
    if (tid < 16) normJ[tid] = norms[j0 + tid];
    if (j0 + 16 < N_ROWS)            // hint next chunk into cache
      __builtin_prefetch(x + (size_t)(j0 + 16) * DIM + tid * 64, 0, 1);
    __syncthreads();

    // ---- GEMM1: S(16x16) = Xi_w (16xD) * Xj^T (Dx16), K-quads over D ----
    v8f S = zero;
#pragma unroll
    for (int kq = 0; kq < 64; ++kq) {
      int d0 = kq * 4 + 2 * half;
      v2f a = *(const v2f*)&Xi[wave][l16][d0];          // A[m=l16][d0,d0+1]
      v2f b;                                            // B[d][n=l16]
      b.x = XjT[d0][l16];
      b.y = XjT[d0 + 1][l16];
      S = wmma4(a, b, S);
    }

    // ---- scale by 1/max(ni*nj, eps), stage to LDS in A-friendly layout --
    float nj = normJ[l16];
#pragma unroll
    for (int r = 0; r < 8; ++r) {
      float sc = __builtin_amdgcn_rcpf(fmaxf(nIm[r] * nj, EPS_CS));
      Ssc[wave][r + 8 * half][l16] = S[r] * sc;         // (m, n) -> LDS
    }

    // ---- GEMM2: acc(16x256) += Ssc(16x16) * Xj(16x256) ------------------
#pragma unroll
    for (int kq = 0; kq < 4; ++kq) {
      v2f a = *(const v2f*)&Ssc[wave][l16][kq * 4 + 2 * half];
#pragma unroll
      for (int t = 0; t < 16; ++t) {
        v2f b = *(const v2f*)&XjT[t * 16 + l16][kq * 4 + 2 * half];
        acc[t] = wmma4(a, b, acc[t]);
      }
    }
  }

  // ---- epilogue: x_agg = acc + x, scatter-add into per-class sums -------
#pragma unroll
  for (int r = 0; r < 8; ++r) {
    int m    = r + 8 * half;
    int gi   = i0 + wave * 16 + m;
    int cidx = cls[gi];
    float* dst = proto_sums + (size_t)cidx * DIM;
#pragma unroll
    for (int t = 0; t < 16; ++t) {
      int col = t * 16 + l16;
      atomicAdd(dst + col, acc[t][r] + Xi[wave][m][col]);
    }
  }
}

// ---------------- kernel 5: normalize prototypes ------------------------
__global__ void k_protonorm(float* __restrict__ proto,
                            const unsigned* __restrict__ counts) {
  int i = blockIdx.x * blockDim.x + threadIdx.x;
  if (i >= N_CLS * DIM) return;
  unsigned c = counts[i >> 8];       // DIM == 256
  float v = proto[i];
  proto[i] = (c > 0u) ? v / (float)c : 0.0f;
}

// ---------------- kernel 6: inter-class difference (1 GB stream) --------
__global__ void k_inter(const float* __restrict__ proto, float* __restrict__ inter) {
  int j = blockIdx.x;
  int i = blockIdx.y;
  int c = threadIdx.x << 2;
  v4f pj = *(const v4f*)(proto + (size_t)j * DIM + c);
  v4f pi = *(const v4f*)(proto + (size_t)i * DIM + c);
  v4f r  = pj - pi;
  __builtin_nontemporal_store(r, (v4f*)(inter + ((size_t)i * N_CLS + j) * DIM + c));
}

// ------------------------------------------------------------------------
extern "C" void kernel_launch(void* const* d_in, const int* in_sizes, int n_in,
                              void* d_out, int out_size, void* d_ws, size_t ws_size,
                              hipStream_t stream) {
  const float* x      = (const float*)d_in[0];   // [4096,256]
  const float* logits = (const float*)d_in[1];   // [4096,1000]

  float* proto = (float*)d_out;                  // [1000,256]
  float* inter = proto + (size_t)N_CLS * DIM;    // [1000,1000,256]

  char* ws = (char*)d_ws;                        // 36 KB used
  float*    norms  = (float*)(ws);               // 4096 f32
  int*      cls    = (int*)(ws + 16384);         // 4096 i32
  unsigned* counts = (unsigned*)(ws + 32768);    // 1000 u32

  k_zero<<<(N_CLS * DIM + N_CLS + 255) / 256, 256, 0, stream>>>(proto, counts);
  k_argmax<<<(N_ROWS * 32) / 256, 256, 0, stream>>>(logits, cls, counts);
  k_rownorm<<<(N_ROWS * 32) / 256, 256, 0, stream>>>(x, norms);
  k_fused_agg<<<N_ROWS / 32, 64, 0, stream>>>(x, norms, cls, proto);
  k_protonorm<<<(N_CLS * DIM + 255) / 256, 256, 0, stream>>>(proto, counts);
  dim3 ig(N_CLS, N_CLS);
  k_inter<<<ig, DIM / 4, 0, stream>>>(proto, inter);
}